// SimGCL_41240275976433
// MI455X (gfx1250) — compile-verified
//
#include <hip/hip_runtime.h>
#include <cstdint>
#include <cstddef>

#define NUM_USERS 200000
#define NUM_ITEMS 100000
#define N_NODES   300000
#define EMBED_DIM 128

#define TILE     1024          // edges staged per LDS tile
#define THREADS  256           // 8 wave32s per block
#define WAVES_PB (THREADS / 32)

// ---- CDNA5 async global->LDS helpers (inline asm: portable across toolchains) ----
__device__ __forceinline__ uint32_t lds_off_of(const void* p) {
    // generic pointer to __shared__ has LDS offset in low 32 bits (aperture form)
    return (uint32_t)(uintptr_t)p;
}

__device__ __forceinline__ void async_load_b128(uint32_t lds_off, const void* gaddr) {
    asm volatile("global_load_async_to_lds_b128 %0, %1, off"
                 :: "v"(lds_off), "v"(gaddr) : "memory");
}

__device__ __forceinline__ void wait_async0() {
    asm volatile("s_wait_asynccnt 0" ::: "memory");
}

__device__ __forceinline__ void atomic_add_f32(float* p, float v) {
    unsafeAtomicAdd(p, v);   // hardware global_atomic_add_f32 (no-return form)
}

// ------------------------------------------------------------------------------
// SpMM: dst[rows[e]] += vals[e] * src[cols[e]]   (wave-per-edge, f32 atomics)
// Edge metadata is double-buffered through LDS with async DMA loads.
// Lane l owns dwords {l, l+32, l+64, l+96} of the 512B row, so every
// atomic/load instruction covers 128B contiguous across the wave.
// SPLIT=true : src is concat(userW, itemW) given as two pointers.
// ------------------------------------------------------------------------------
template <bool SPLIT>
__device__ __forceinline__
void process_edge(float v, int r, int c, int lane,
                  const float* __restrict__ srcU,
                  const float* __restrict__ srcI,
                  float* __restrict__ dst)
{
    const float* srcRow;
    if (SPLIT) {
        srcRow = (c < NUM_USERS)
                   ? (srcU + (size_t)c * EMBED_DIM)
                   : (srcI + (size_t)(c - NUM_USERS) * EMBED_DIM);
    } else {
        srcRow = srcU + (size_t)c * EMBED_DIM;
    }
    const float* s = srcRow + lane;
    float x0 = s[0];
    float x1 = s[32];
    float x2 = s[64];
    float x3 = s[96];
    float* o = dst + (size_t)r * EMBED_DIM + lane;
    atomic_add_f32(o +  0, v * x0);
    atomic_add_f32(o + 32, v * x1);
    atomic_add_f32(o + 64, v * x2);
    atomic_add_f32(o + 96, v * x3);
}

template <bool SPLIT>
__global__ __launch_bounds__(THREADS)
void spmm_async_kernel(const float* __restrict__ vals,
                       const int*   __restrict__ rows,
                       const int*   __restrict__ cols,
                       const float* __restrict__ srcU,
                       const float* __restrict__ srcI,
                       float*       __restrict__ dst,
                       int nEdges)
{
    __shared__ float s_val[2][TILE];
    __shared__ int   s_row[2][TILE];
    __shared__ int   s_col[2][TILE];

    const int tid  = threadIdx.x;
    const int lane = tid & 31;
    const int wib  = tid >> 5;
    const int nTiles = (nEdges + TILE - 1) / TILE;

    auto issue_tile = [&](int tile, int buf) {
        // each of 256 threads async-loads 16B (4 edges) of each array
        int base = tile * TILE + tid * 4;
        if (base < nEdges) {   // nEdges % 4 == 0, so base+3 stays in range
            async_load_b128(lds_off_of(&s_val[buf][tid * 4]), vals + base);
            async_load_b128(lds_off_of(&s_row[buf][tid * 4]), rows + base);
            async_load_b128(lds_off_of(&s_col[buf][tid * 4]), cols + base);
        }
    };

    int t = (int)blockIdx.x;
    if (t >= nTiles) return;

    int buf = 0;
    issue_tile(t, buf);

    for (; t < nTiles; t += (int)gridDim.x) {
        wait_async0();        // this wave's async portion of current tile done
        __syncthreads();      // all waves' portions done -> tile is in LDS

        int nt = t + (int)gridDim.x;
        if (nt < nTiles) issue_tile(nt, buf ^ 1);   // overlap DMA with compute

        const int base = t * TILE;
        const int cnt  = (nEdges - base) < TILE ? (nEdges - base) : TILE;

        int i = wib;
        // unrolled by 2: two edges in flight to hide LDS + gather latency
        for (; i + WAVES_PB < cnt; i += 2 * WAVES_PB) {
            float v0 = s_val[buf][i];
            int   r0 = s_row[buf][i];
            int   c0 = s_col[buf][i];
            float v1 = s_val[buf][i + WAVES_PB];
            int   r1 = s_row[buf][i + WAVES_PB];
            int   c1 = s_col[buf][i + WAVES_PB];
            v0 = __int_as_float(__builtin_amdgcn_readfirstlane(__float_as_int(v0)));
            r0 = __builtin_amdgcn_readfirstlane(r0);
            c0 = __builtin_amdgcn_readfirstlane(c0);
            v1 = __int_as_float(__builtin_amdgcn_readfirstlane(__float_as_int(v1)));
            r1 = __builtin_amdgcn_readfirstlane(r1);
            c1 = __builtin_amdgcn_readfirstlane(c1);
            process_edge<SPLIT>(v0, r0, c0, lane, srcU, srcI, dst);
            process_edge<SPLIT>(v1, r1, c1, lane, srcU, srcI, dst);
        }
        for (; i < cnt; i += WAVES_PB) {
            float v = s_val[buf][i];
            int   r = s_row[buf][i];
            int   c = s_col[buf][i];
            v = __int_as_float(__builtin_amdgcn_readfirstlane(__float_as_int(v)));
            r = __builtin_amdgcn_readfirstlane(r);
            c = __builtin_amdgcn_readfirstlane(c);
            process_edge<SPLIT>(v, r, c, lane, srcU, srcI, dst);
        }
        buf ^= 1;
    }
}

// d_out = ego (concat of user/item weights); ws = 0
__global__ __launch_bounds__(THREADS)
void init_kernel(const float* __restrict__ uw, const float* __restrict__ iw,
                 float* __restrict__ out, float* __restrict__ ws)
{
    const size_t n4   = (size_t)N_NODES * EMBED_DIM / 4;
    const size_t usr4 = (size_t)NUM_USERS * EMBED_DIM / 4;
    const size_t stride = (size_t)gridDim.x * blockDim.x;
    for (size_t i = (size_t)blockIdx.x * blockDim.x + threadIdx.x; i < n4; i += stride) {
        float4 v = (i < usr4) ? reinterpret_cast<const float4*>(uw)[i]
                              : reinterpret_cast<const float4*>(iw)[i - usr4];
        reinterpret_cast<float4*>(out)[i] = v;
        reinterpret_cast<float4*>(ws)[i]  = make_float4(0.f, 0.f, 0.f, 0.f);
    }
}

// d_out = (d_out + ws) / 3       [d_out holds ego + e2, ws holds e1]
__global__ __launch_bounds__(THREADS)
void finalize_kernel(float* __restrict__ out, const float* __restrict__ ws)
{
    const size_t n4 = (size_t)N_NODES * EMBED_DIM / 4;
    const float  s  = 1.0f / 3.0f;
    const size_t stride = (size_t)gridDim.x * blockDim.x;
    for (size_t i = (size_t)blockIdx.x * blockDim.x + threadIdx.x; i < n4; i += stride) {
        float4 a = reinterpret_cast<float4*>(out)[i];
        float4 b = reinterpret_cast<const float4*>(ws)[i];
        a.x = (a.x + b.x) * s;
        a.y = (a.y + b.y) * s;
        a.z = (a.z + b.z) * s;
        a.w = (a.w + b.w) * s;
        reinterpret_cast<float4*>(out)[i] = a;
    }
}

extern "C" void kernel_launch(void* const* d_in, const int* in_sizes, int n_in,
                              void* d_out, int out_size, void* d_ws, size_t ws_size,
                              hipStream_t stream)
{
    const float* uw = (const float*)d_in[0];   // user_weight  [200000 x 128]
    const float* iw = (const float*)d_in[1];   // item_weight  [100000 x 128]
    const float* ev = (const float*)d_in[2];   // edge_vals    [9.6M]
    const int*   er = (const int*)  d_in[3];   // edge_rows
    const int*   ec = (const int*)  d_in[4];   // edge_cols
    float* out = (float*)d_out;                // final embeddings [300000 x 128]
    float* e1  = (float*)d_ws;                 // layer-1 output, 153.6 MB scratch

    const int nE = in_sizes[2];

    // 1) out = ego, e1 = 0
    init_kernel<<<2048, THREADS, 0, stream>>>(uw, iw, out, e1);

    // 2) e1 = spmm(ego)
    int nTiles = (nE + TILE - 1) / TILE;
    int blocks = nTiles < 1024 ? nTiles : 1024;
    spmm_async_kernel<true><<<blocks, THREADS, 0, stream>>>(ev, er, ec, uw, iw, e1, nE);

    // 3) out += spmm(e1)   (accumulate e2 directly onto ego)
    spmm_async_kernel<false><<<blocks, THREADS, 0, stream>>>(ev, er, ec, e1, nullptr, out, nE);

    // 4) out = (out + e1) / 3
    finalize_kernel<<<2048, THREADS, 0, stream>>>(out, e1);
}